// GCN_24721831756423
// MI455X (gfx1250) — compile-verified
//
#include <hip/hip_runtime.h>
#include <hip/hip_bf16.h>

// GCN 4-layer forward for MI455X (gfx1250, wave32).
//   L1 GEMM x@W1 via V_WMMA_F32_16X16X4_F32 (f32 precision, memory-bound anyway)
//   edge scatter via f32 atomics (agg buffers are L2-resident: <= 6.4 MB)
//   tiny layer GEMMs fused with relu(agg+b) epilogue (scalar VALU)
// Workspace: two [N,16] f32 ping-pong buffers (12.8 MB total).

typedef __attribute__((ext_vector_type(2))) float v2f;
typedef __attribute__((ext_vector_type(8))) float v8f;

#define NFEAT 512
#define PADK  516   // LDS k-stride: 516 % 64 == 4 -> 32 lanes' b64 reads hit all 64 banks

// ---------------------------------------------------------------------------
// Layer-1 GEMM: out[n,0:8] = x[n,:] @ W1[512,8], one 16-row tile per wave.
// A layout (16x4 f32): lane l in [0,15]: row=tile*16+l, holds {K, K+1};
//                      lane l in [16,31]: same rows, holds {K+2, K+3}.
// B layout (4x16 f32): lane n in [0,15]: col n, holds {W[K][n], W[K+1][n]};
//                      lanes 16..31: col n-16, holds {W[K+2][n], W[K+3][n]}.
// W1 is zero-padded to 16 columns in LDS so the B load is branchless
// (no per-iteration EXEC save/restore, tight load+wmma loop).
// D layout: VGPR r -> row tile*16 + r + (lane>=16 ? 8 : 0), col = lane&15.
// ---------------------------------------------------------------------------
__global__ void gcn_gemm1_wmma(const float* __restrict__ x,
                               const float* __restrict__ W1,
                               float* __restrict__ out, int nTiles) {
  __shared__ float ldsW[16 * PADK];          // transposed + padded: ldsW[col][k]
  for (int i = threadIdx.x; i < NFEAT * 16; i += blockDim.x) {
    int k = i >> 4, c = i & 15;
    ldsW[c * PADK + k] = (c < 8) ? W1[k * 8 + c] : 0.0f;
  }
  __syncthreads();

  const int wavesPerBlock = blockDim.x >> 5;
  const int wave = threadIdx.x >> 5;
  const int lane = threadIdx.x & 31;
  const int tile = blockIdx.x * wavesPerBlock + wave;
  if (tile >= nTiles) return;                // uniform per wave

  const int half = lane >> 4;                // 0: K,K+1   1: K+2,K+3
  const int l15  = lane & 15;
  const float* arow = x + (size_t)(tile * 16 + l15) * NFEAT + half * 2;
  const float* brow = &ldsW[l15 * PADK + half * 2];

  __builtin_prefetch(arow + 256, 0, 3);      // pull row tail toward WGP (global_prefetch_b8)

  const int NK = NFEAT / 4;                  // 128 k-steps
  v2f a = *(const v2f*)(arow);
  v2f b = *(const v2f*)(brow);
  v8f acc = {};
  #pragma unroll 4
  for (int k = 0; k < NK - 1; ++k) {
    v2f an = *(const v2f*)(arow + (k + 1) * 4);   // issue next loads before the
    v2f bn = *(const v2f*)(brow + (k + 1) * 4);   // wmma so latency overlaps
    acc = __builtin_amdgcn_wmma_f32_16x16x4_f32(
        false, a, false, b, (short)0, acc, false, false);
    a = an; b = bn;
  }
  acc = __builtin_amdgcn_wmma_f32_16x16x4_f32(
      false, a, false, b, (short)0, acc, false, false);

  float* orow = out + ((size_t)tile * 16 + half * 8) * 16 + l15;
  #pragma unroll
  for (int r = 0; r < 8; ++r) orow[r * 16] = acc[r];
}

// ---------------------------------------------------------------------------
// Edge scatter: agg[dst,f] += h[src,f] * w   (f32 atomics, L2-resident agg)
// ---------------------------------------------------------------------------
template <int F>
__global__ void gcn_scatter(const int* __restrict__ src, const int* __restrict__ dst,
                            const float* __restrict__ w, const float* __restrict__ h,
                            float* __restrict__ agg, int E) {
  int e = blockIdx.x * blockDim.x + threadIdx.x;
  if (e >= E) return;
  int s = src[e], d = dst[e];
  float ww = w[e];
  const float* hr = h + (size_t)s * 16;
  float* ar = agg + (size_t)d * 16;
  float m[F];
  #pragma unroll
  for (int f = 0; f < F; ++f) m[f] = hr[f] * ww;   // b128-merged gather
  #pragma unroll
  for (int f = 0; f < F; ++f) atomicAdd(&ar[f], m[f]);
}

// ---------------------------------------------------------------------------
// Per-node: h = relu(agg + b); out = h @ W  (tiny, node-invariant W -> s_loads)
// ---------------------------------------------------------------------------
template <int FIN, int FOUT>
__global__ void gcn_node_mlp(const float* __restrict__ agg, const float* __restrict__ b,
                             const float* __restrict__ W, float* __restrict__ out, int N) {
  int n = blockIdx.x * blockDim.x + threadIdx.x;
  if (n >= N) return;
  const float* ar = agg + (size_t)n * 16;
  float h[FIN];
  #pragma unroll
  for (int i = 0; i < FIN; ++i) {
    float v = ar[i] + b[i];
    h[i] = v > 0.f ? v : 0.f;
  }
  float o[FOUT];
  #pragma unroll
  for (int j = 0; j < FOUT; ++j) o[j] = 0.f;
  #pragma unroll
  for (int i = 0; i < FIN; ++i)
    #pragma unroll
    for (int j = 0; j < FOUT; ++j) o[j] = fmaf(h[i], W[i * FOUT + j], o[j]);
  float* orow = out + (size_t)n * 16;
  #pragma unroll
  for (int j = 0; j < FOUT; ++j) orow[j] = o[j];
}

// ---------------------------------------------------------------------------
// Final: out = log_softmax(agg + b4) over 10 classes
// ---------------------------------------------------------------------------
__global__ void gcn_logsoftmax(const float* __restrict__ agg, const float* __restrict__ b,
                               float* __restrict__ out, int N) {
  int n = blockIdx.x * blockDim.x + threadIdx.x;
  if (n >= N) return;
  const float* ar = agg + (size_t)n * 16;
  float v[10], m = -3.0e38f;
  #pragma unroll
  for (int c = 0; c < 10; ++c) { v[c] = ar[c] + b[c]; m = fmaxf(m, v[c]); }
  float s = 0.f;
  #pragma unroll
  for (int c = 0; c < 10; ++c) s += __expf(v[c] - m);
  float ls = __logf(s);
  float* orow = out + (size_t)n * 10;
  #pragma unroll
  for (int c = 0; c < 10; ++c) orow[c] = v[c] - m - ls;
}

extern "C" void kernel_launch(void* const* d_in, const int* in_sizes, int n_in,
                              void* d_out, int out_size, void* d_ws, size_t ws_size,
                              hipStream_t stream) {
  const float* x  = (const float*)d_in[0];
  const int*   ei = (const int*)d_in[1];      // edge_index [2,E]
  const float* ew = (const float*)d_in[2];
  const float* W1 = (const float*)d_in[3];
  const float* b1 = (const float*)d_in[4];
  const float* W2 = (const float*)d_in[5];
  const float* b2 = (const float*)d_in[6];
  const float* W3 = (const float*)d_in[7];
  const float* b3 = (const float*)d_in[8];
  const float* W4 = (const float*)d_in[9];
  const float* b4 = (const float*)d_in[10];

  const int N = in_sizes[0] / NFEAT;          // 100000 (multiple of 16)
  const int E = in_sizes[1] / 2;              // 3200000
  const int* src = ei;
  const int* dst = ei + E;

  float* bufA = (float*)d_ws;                 // [N,16]
  float* bufB = bufA + (size_t)N * 16;        // [N,16]
  const size_t rowBytes = (size_t)N * 16 * sizeof(float);

  const int nTiles = N / 16;
  const int WPB = 256 / 32;                   // waves per block

  // Layer 1: GEMM (WMMA) -> bufA[:,0:8]
  gcn_gemm1_wmma<<<(nTiles + WPB - 1) / WPB, 256, 0, stream>>>(x, W1, bufA, nTiles);
  hipMemsetAsync(bufB, 0, rowBytes, stream);
  gcn_scatter<8><<<(E + 255) / 256, 256, 0, stream>>>(src, dst, ew, bufA, bufB, E);

  // Layer 2: relu(agg1+b1) @ W2 -> bufA[:,0:16]
  gcn_node_mlp<8, 16><<<(N + 255) / 256, 256, 0, stream>>>(bufB, b1, W2, bufA, N);
  hipMemsetAsync(bufB, 0, rowBytes, stream);
  gcn_scatter<16><<<(E + 255) / 256, 256, 0, stream>>>(src, dst, ew, bufA, bufB, E);

  // Layer 3: relu(agg2+b2) @ W3 -> bufA[:,0:8]
  gcn_node_mlp<16, 8><<<(N + 255) / 256, 256, 0, stream>>>(bufB, b2, W3, bufA, N);
  hipMemsetAsync(bufB, 0, rowBytes, stream);
  gcn_scatter<8><<<(E + 255) / 256, 256, 0, stream>>>(src, dst, ew, bufA, bufB, E);

  // Layer 4: relu(agg3+b3) @ W4 -> bufA[:,0:10]
  gcn_node_mlp<8, 10><<<(N + 255) / 256, 256, 0, stream>>>(bufB, b3, W4, bufA, N);
  hipMemsetAsync(bufB, 0, rowBytes, stream);
  gcn_scatter<10><<<(E + 255) / 256, 256, 0, stream>>>(src, dst, ew, bufA, bufB, E);

  // log_softmax(agg4 + b4) -> d_out [N,10]
  gcn_logsoftmax<<<(N + 255) / 256, 256, 0, stream>>>(bufB, b4, (float*)d_out, N);
}